// HRCDC_24309514895386
// MI455X (gfx1250) — compile-verified
//
#include <hip/hip_runtime.h>
#include <math.h>

// Problem constants (match reference)
constexpr int Nn  = 100000;
constexpr int Dd  = 128;
constexpr int Rr  = 4;
constexpr int Ee  = 150000;
constexpr int Hh  = 8;
constexpr int DKk = 16;

// GEMM tiling: 5 row-tiles (80 rows) per block; 100000 / 80 = 1250 exactly.
constexpr int TILES = 5;
constexpr int ROWS  = 16 * TILES;

typedef __attribute__((ext_vector_type(2))) float v2f;
typedef __attribute__((ext_vector_type(8))) float v8f;

__device__ __forceinline__ v8f wmma_f32_16x16x4(v2f a, v2f b, v8f c) {
  // (neg_a, A, neg_b, B, c_mod, C, reuse_a, reuse_b)
  return __builtin_amdgcn_wmma_f32_16x16x4_f32(
      false, a, false, b, (short)0, c, false, false);
}

__device__ __forceinline__ void atomicMaxFloat(float* addr, float val) {
  if (val >= 0.f) atomicMax((int*)addr, __float_as_int(val));
  else            atomicMin((unsigned int*)addr, (unsigned int)__float_as_int(val));
}

// ---------------------------------------------------------------------------
// Multi-tile WMMA accumulate: one B fragment (b64/lane) feeds TILES WMMAs.
// A layout (ISA 16x4 f32): vgpr0 = K{0|2}, vgpr1 = K{1|3}; B mirrors A.
// ---------------------------------------------------------------------------
__device__ __forceinline__ void gemm_tiles(const float (&xs)[ROWS][132],
                                           const float* __restrict__ W,
                                           const float* __restrict__ bias,
                                           float* __restrict__ O, int row0,
                                           int j0, int lane) {
  const int mrow = lane & 15;
  const int kof  = (lane < 16) ? 0 : 2;
  const float* wrow = W + (size_t)(j0 + mrow) * Dd;  // B[k][n] = W[j0+n][k]

  v8f c[TILES];
#pragma unroll
  for (int t = 0; t < TILES; ++t) c[t] = (v8f){};

  for (int kk = 0; kk < Dd; kk += 4) {
    v2f b;
    b[0] = wrow[kk + kof];
    b[1] = wrow[kk + kof + 1];
#pragma unroll
    for (int t = 0; t < TILES; ++t) {
      v2f a;
      a[0] = xs[t * 16 + mrow][kk + kof];
      a[1] = xs[t * 16 + mrow][kk + kof + 1];
      c[t] = wmma_f32_16x16x4(a, b, c[t]);
    }
  }

  const float bv = bias[j0 + mrow];
#pragma unroll
  for (int t = 0; t < TILES; ++t) {
    const int rbase = row0 + t * 16 + ((lane < 16) ? 0 : 8);
#pragma unroll
    for (int i = 0; i < 8; ++i)
      O[(size_t)(rbase + i) * Dd + j0 + mrow] = c[t][i] + bv;
  }
}

// ---------------------------------------------------------------------------
// 0) workspace init: mseg = -inf, sseg = 0, T = 0
// ---------------------------------------------------------------------------
__global__ void init_ws(float* __restrict__ mseg, float* __restrict__ sseg,
                        float* __restrict__ T) {
  const long nmh = (long)Rr * Nn * Hh;
  const long ntd = (long)Rr * Nn * Dd;
  const long total = 2 * nmh + ntd;
  for (long i = (long)blockIdx.x * blockDim.x + threadIdx.x; i < total;
       i += (long)gridDim.x * blockDim.x) {
    if (i < nmh)          mseg[i] = -INFINITY;
    else if (i < 2 * nmh) sseg[i - nmh] = 0.f;
    else                  T[i - 2 * nmh] = 0.f;
  }
}

// ---------------------------------------------------------------------------
// 1) fused K/Q/V projection. 80 x-rows staged in LDS per block; 8 waves own
//    16-column slices; each B fragment is reused across 5 row tiles.
// ---------------------------------------------------------------------------
__global__ __launch_bounds__(256) void proj_kqv(
    const float* __restrict__ x,
    const float* __restrict__ Wk, const float* __restrict__ bk,
    const float* __restrict__ Wq, const float* __restrict__ bq,
    const float* __restrict__ Wv, const float* __restrict__ bv,
    float* __restrict__ Kp, float* __restrict__ Qp, float* __restrict__ Vp) {
  __shared__ float xs[ROWS][132];  // 80*132*4 = 42.2 KB; pad spreads banks
  const int row0 = blockIdx.x * ROWS;
  const int tid  = threadIdx.x;

  for (int i = tid; i < ROWS * 128; i += 256) {
    const int rr = i >> 7, cc = i & 127;
    xs[rr][cc] = x[(size_t)(row0 + rr) * Dd + cc];
  }
  __syncthreads();

  const int wave = tid >> 5, lane = tid & 31;
  const int j0   = wave * 16;

  gemm_tiles(xs, Wk, bk, Kp, row0, j0, lane);
  gemm_tiles(xs, Wq, bq, Qp, row0, j0, lane);
  gemm_tiles(xs, Wv, bv, Vp, row0, j0, lane);
}

// ---------------------------------------------------------------------------
// 2) edge attention logits + segment max
// ---------------------------------------------------------------------------
__global__ __launch_bounds__(256) void edge_att(
    const float* __restrict__ Kp, const float* __restrict__ Qp,
    const int* __restrict__ esrc, const int* __restrict__ edst,
    const float* __restrict__ rel_att, const float* __restrict__ rel_pri,
    float* __restrict__ att, float* __restrict__ mseg) {
  __shared__ float rA[Hh * DKk * DKk];  // 2048 floats
  const int r   = blockIdx.y;
  const int tid = threadIdx.x;
  for (int i = tid; i < Hh * DKk * DKk; i += 256)
    rA[i] = rel_att[(size_t)r * Hh * DKk * DKk + i];
  __syncthreads();

  const long idx = (long)blockIdx.x * 256 + tid;
  if (idx >= (long)Ee * Hh) return;
  const int e = (int)(idx >> 3), h = (int)(idx & 7);
  const int src = esrc[(size_t)r * Ee + e];
  const int dst = edst[(size_t)r * Ee + e];

  const float4* kp = (const float4*)(Kp + (size_t)src * Dd + h * DKk);
  const float4* qp = (const float4*)(Qp + (size_t)dst * Dd + h * DKk);
  float kv[16], qv[16];
#pragma unroll
  for (int i = 0; i < 4; ++i) {
    const float4 k4 = kp[i], q4 = qp[i];
    kv[4 * i + 0] = k4.x; kv[4 * i + 1] = k4.y;
    kv[4 * i + 2] = k4.z; kv[4 * i + 3] = k4.w;
    qv[4 * i + 0] = q4.x; qv[4 * i + 1] = q4.y;
    qv[4 * i + 2] = q4.z; qv[4 * i + 3] = q4.w;
  }

  const float* A = &rA[h * 256];
  float acc = 0.f;
#pragma unroll
  for (int f = 0; f < 16; ++f) {
    float kf = 0.f;
#pragma unroll
    for (int d = 0; d < 16; ++d) kf = fmaf(kv[d], A[d * 16 + f], kf);
    acc = fmaf(qv[f], kf, acc);
  }
  acc *= rel_pri[r * Hh + h] * 0.25f;  // /sqrt(16)

  att[((size_t)r * Ee + e) * Hh + h] = acc;
  atomicMaxFloat(&mseg[((size_t)r * Nn + dst) * Hh + h], acc);
}

// ---------------------------------------------------------------------------
// 3) exp / segment sum / weighted message scatter
// ---------------------------------------------------------------------------
__global__ __launch_bounds__(256) void edge_msg(
    const float* __restrict__ Vp,
    const int* __restrict__ esrc, const int* __restrict__ edst,
    const float* __restrict__ rel_msg, const float* __restrict__ att,
    const float* __restrict__ mseg, float* __restrict__ sseg,
    float* __restrict__ T) {
  __shared__ float rM[Hh * DKk * DKk];
  const int r   = blockIdx.y;
  const int tid = threadIdx.x;
  for (int i = tid; i < Hh * DKk * DKk; i += 256)
    rM[i] = rel_msg[(size_t)r * Hh * DKk * DKk + i];
  __syncthreads();

  const long idx = (long)blockIdx.x * 256 + tid;
  if (idx >= (long)Ee * Hh) return;
  const int e = (int)(idx >> 3), h = (int)(idx & 7);
  const int src = esrc[(size_t)r * Ee + e];
  const int dst = edst[(size_t)r * Ee + e];
  const size_t seg = (size_t)r * Nn + dst;

  const float a =
      __expf(att[((size_t)r * Ee + e) * Hh + h] - mseg[seg * Hh + h]);
  atomicAdd(&sseg[seg * Hh + h], a);

  const float4* vp = (const float4*)(Vp + (size_t)src * Dd + h * DKk);
  float vv[16];
#pragma unroll
  for (int i = 0; i < 4; ++i) {
    const float4 v4 = vp[i];
    vv[4 * i + 0] = v4.x; vv[4 * i + 1] = v4.y;
    vv[4 * i + 2] = v4.z; vv[4 * i + 3] = v4.w;
  }

  const float* M = &rM[h * 256];
  float* Tp = T + seg * Dd + h * DKk;
#pragma unroll
  for (int f = 0; f < 16; ++f) {
    float mf = 0.f;
#pragma unroll
    for (int d = 0; d < 16; ++d) mf = fmaf(vv[d], M[d * 16 + f], mf);
    atomicAdd(&Tp[f], a * mf);
  }
}

// ---------------------------------------------------------------------------
// 4) normalize by segment sum + mean over relations, in place into T[r=0]
// ---------------------------------------------------------------------------
__global__ void mean_norm(const float* __restrict__ sseg, float* __restrict__ T) {
  const long idx = (long)blockIdx.x * blockDim.x + threadIdx.x;
  if (idx >= (long)Nn * Dd) return;
  const int n = (int)(idx / Dd), d = (int)(idx % Dd), h = d >> 4;
  float acc = 0.f;
#pragma unroll
  for (int r = 0; r < Rr; ++r) {
    const size_t seg = (size_t)r * Nn + n;
    const float s  = sseg[seg * Hh + h];
    const float tv = T[seg * Dd + d];
    acc += (s > 0.f) ? tv / s : 0.f;
  }
  T[(size_t)n * Dd + d] = acc * (1.f / Rr);
}

// ---------------------------------------------------------------------------
// 5) output GEMM (Tm @ Wa^T + ba) fused with sigmoid skip blend
// ---------------------------------------------------------------------------
__global__ __launch_bounds__(256) void out_gemm(
    const float* __restrict__ Tm, const float* __restrict__ Wa,
    const float* __restrict__ ba, const float* __restrict__ x,
    const float* __restrict__ skip, float* __restrict__ out) {
  __shared__ float ts[ROWS][132];
  const int row0 = blockIdx.x * ROWS;
  const int tid  = threadIdx.x;
  for (int i = tid; i < ROWS * 128; i += 256) {
    const int rr = i >> 7, cc = i & 127;
    ts[rr][cc] = Tm[(size_t)(row0 + rr) * Dd + cc];
  }
  __syncthreads();

  const float alpha = 1.f / (1.f + __expf(-skip[0]));
  const int wave = tid >> 5, lane = tid & 31;
  const int j0   = wave * 16;
  const int mrow = lane & 15;
  const int kof  = (lane < 16) ? 0 : 2;
  const float* wrow = Wa + (size_t)(j0 + mrow) * Dd;

  v8f c[TILES];
#pragma unroll
  for (int t = 0; t < TILES; ++t) c[t] = (v8f){};

  for (int kk = 0; kk < Dd; kk += 4) {
    v2f b;
    b[0] = wrow[kk + kof];
    b[1] = wrow[kk + kof + 1];
#pragma unroll
    for (int t = 0; t < TILES; ++t) {
      v2f a;
      a[0] = ts[t * 16 + mrow][kk + kof];
      a[1] = ts[t * 16 + mrow][kk + kof + 1];
      c[t] = wmma_f32_16x16x4(a, b, c[t]);
    }
  }

  const float bv = ba[j0 + mrow];
#pragma unroll
  for (int t = 0; t < TILES; ++t) {
    const int rbase = row0 + t * 16 + ((lane < 16) ? 0 : 8);
#pragma unroll
    for (int i = 0; i < 8; ++i) {
      const size_t off = (size_t)(rbase + i) * Dd + j0 + mrow;
      out[off] = (c[t][i] + bv) * alpha + x[off] * (1.f - alpha);
    }
  }
}

// ---------------------------------------------------------------------------
extern "C" void kernel_launch(void* const* d_in, const int* in_sizes, int n_in,
                              void* d_out, int out_size, void* d_ws,
                              size_t ws_size, hipStream_t stream) {
  const float* x       = (const float*)d_in[0];
  const int*   esrc    = (const int*)d_in[1];
  const int*   edst    = (const int*)d_in[2];
  const float* Wk      = (const float*)d_in[3];
  const float* bk      = (const float*)d_in[4];
  const float* Wq      = (const float*)d_in[5];
  const float* bq      = (const float*)d_in[6];
  const float* Wv      = (const float*)d_in[7];
  const float* bv      = (const float*)d_in[8];
  const float* Wa      = (const float*)d_in[9];
  const float* ba      = (const float*)d_in[10];
  const float* rel_att = (const float*)d_in[11];
  const float* rel_msg = (const float*)d_in[12];
  const float* rel_pri = (const float*)d_in[13];
  const float* skip    = (const float*)d_in[14];
  float* out = (float*)d_out;

  // workspace layout (floats): K,Q,V | att | mseg | sseg | T
  float* ws   = (float*)d_ws;
  float* Kp   = ws;
  float* Qp   = Kp + (size_t)Nn * Dd;
  float* Vp   = Qp + (size_t)Nn * Dd;
  float* att  = Vp + (size_t)Nn * Dd;
  float* mseg = att + (size_t)Rr * Ee * Hh;
  float* sseg = mseg + (size_t)Rr * Nn * Hh;
  float* T    = sseg + (size_t)Rr * Nn * Hh;

  init_ws<<<2048, 256, 0, stream>>>(mseg, sseg, T);

  proj_kqv<<<Nn / ROWS, 256, 0, stream>>>(x, Wk, bk, Wq, bq, Wv, bv,
                                          Kp, Qp, Vp);

  dim3 ge(((long)Ee * Hh + 255) / 256, Rr);
  edge_att<<<ge, 256, 0, stream>>>(Kp, Qp, esrc, edst, rel_att, rel_pri, att, mseg);
  edge_msg<<<ge, 256, 0, stream>>>(Vp, esrc, edst, rel_msg, att, mseg, sseg, T);

  mean_norm<<<((long)Nn * Dd + 255) / 256, 256, 0, stream>>>(sseg, T);

  out_gemm<<<Nn / ROWS, 256, 0, stream>>>(T, Wa, ba, x, skip, out);
}